// TripletMiningLoss_64939905515573
// MI455X (gfx1250) — compile-verified
//
#include <hip/hip_runtime.h>

typedef float v2f __attribute__((ext_vector_type(2)));
typedef float v8f __attribute__((ext_vector_type(8)));

#define NB      4096          // batch
#define ND      256           // embedding dim
#define NKC     64            // K-chunks of 4 (ND/4)
#define TMARGIN 0.3f
#define TEPS    1e-6f
#define DSTRIDE 4100          // LDS row stride (bank-conflict pad: 4100 % 64 == 4)

// ---------------- kernel 1: row squared norms (wave per row) ----------------
__global__ __launch_bounds__(256) void sq_norm_kernel(const float* __restrict__ emb,
                                                      float* __restrict__ sq) {
    const int wave = threadIdx.x >> 5;
    const int lane = threadIdx.x & 31;
    const int row  = blockIdx.x * 8 + wave;           // 512 blocks * 8 waves = 4096
    const float4* p = (const float4*)(emb + row * ND) + lane * 2;
    float4 x0 = p[0], x1 = p[1];
    float s = x0.x*x0.x + x0.y*x0.y + x0.z*x0.z + x0.w*x0.w
            + x1.x*x1.x + x1.y*x1.y + x1.z*x1.z + x1.w*x1.w;
    #pragma unroll
    for (int m = 16; m >= 1; m >>= 1) s += __shfl_xor(s, m, 32);
    if (lane == 0) sq[row] = s;
}

// ------- kernel 2: fused distance-stripe (WMMA) + mining + triplet loss -------
__global__ __launch_bounds__(256) void triplet_kernel(const float* __restrict__ emb,
                                                      const int*   __restrict__ labels,
                                                      const float* __restrict__ sq,
                                                      float*       __restrict__ partial) {
    extern __shared__ float smem[];                   // dist[16*DSTRIDE] + lsum[16] + lcnt[16]
    float* dist = smem;
    float* lsum = smem + 16 * DSTRIDE;
    float* lcnt = lsum + 16;

    const int t    = threadIdx.x;
    const int wave = t >> 5;
    const int lane = t & 31;
    const int n    = lane & 15;                       // row-in-tile / col-in-tile
    const int h    = lane >> 4;                       // k-half selector per WMMA layout
    const int i0   = blockIdx.x * 16;                 // anchor block base

    // ---- A fragments: 16 anchor rows x 256 dims, layout per 16x16x4 f32 A spec ----
    v2f a[NKC];
    const float* arow = emb + (i0 + n) * ND;
    #pragma unroll
    for (int kc = 0; kc < NKC; ++kc)
        a[kc] = *(const v2f*)(arow + kc * 4 + 2 * h);

    float sqi[8];
    #pragma unroll
    for (int v = 0; v < 8; ++v) sqi[v] = sq[i0 + v + 8 * h];   // C row = v + 8h

    // ---- each wave computes a 16 x 512 slice of the distance stripe ----
    const int jbase = wave * 512;
    for (int jt = 0; jt < 32; ++jt) {
        const int j0 = jbase + jt * 16;
        const float* brow = emb + (j0 + n) * ND;      // B = E^T tile: B[k][n] = E[j0+n][k]
        if (jt + 1 < 32) __builtin_prefetch(brow + 16 * ND, 0, 0);   // global_prefetch_b8
        v8f acc = {};
        #pragma unroll
        for (int kc = 0; kc < NKC; ++kc) {
            v2f b = *(const v2f*)(brow + kc * 4 + 2 * h);
            acc = __builtin_amdgcn_wmma_f32_16x16x4_f32(false, a[kc], false, b,
                                                        (short)0, acc, false, false);
        }
        const float sqj = sq[j0 + n];
        #pragma unroll
        for (int v = 0; v < 8; ++v) {
            float d2 = sqi[v] + sqj - 2.0f * acc[v];
            dist[(v + 8 * h) * DSTRIDE + j0 + n] = sqrtf(fmaxf(d2, 0.0f));
        }
    }
    __syncthreads();

    // ---- mining: 16 lanes per anchor, strided column scan ----
    const int a_id = t >> 4;                          // anchor 0..15
    const int c0   = t & 15;
    const int ga   = i0 + a_id;                       // global anchor index
    const int la   = labels[ga];
    const float* drow = dist + a_id * DSTRIDE;

    // pass 1: hardest positive (first-argmax) + hardest negative (first-argmin)
    float apv = -__builtin_inff(); int api = NB;
    float anv =  __builtin_inff(); int ani = NB;
    for (int c = c0; c < NB; c += 16) {
        const float d   = drow[c];
        const bool same = (labels[c] == la);
        if (same && (c != ga) && d > apv) { apv = d; api = c; }
        if (!same && d < anv)             { anv = d; ani = c; }
    }
    #pragma unroll
    for (int m = 1; m < 16; m <<= 1) {
        float ov = __shfl_xor(apv, m, 16); int oi = __shfl_xor(api, m, 16);
        if (ov > apv || (ov == apv && oi < api)) { apv = ov; api = oi; }
        ov = __shfl_xor(anv, m, 16); oi = __shfl_xor(ani, m, 16);
        if (ov < anv || (ov == anv && oi < ani)) { anv = ov; ani = oi; }
    }

    // pass 2: first semi-hard negative: ap < d < ap + margin
    const float apd = apv;
    int semi = NB;
    for (int c = c0; c < NB; c += 16) {
        const float d = drow[c];
        if ((labels[c] != la) && d > apd && d < apd + TMARGIN && c < semi) semi = c;
    }
    #pragma unroll
    for (int m = 1; m < 16; m <<= 1) {
        int oi = __shfl_xor(semi, m, 16);
        if (oi < semi) semi = oi;
    }

    // ---- exact triplet loss for selected indices (torch adds EPS inside the norm) ----
    const bool valid = (api < NB) && (ani < NB);
    const int  negi  = (semi < NB) ? semi : ani;
    const int  hp    = (api  < NB) ? api  : 0;        // clamp (contribution masked anyway)
    const int  ng    = (negi < NB) ? negi : 0;

    const float* ea = emb + ga * ND + c0 * 16;
    const float* ep = emb + hp * ND + c0 * 16;
    const float* en = emb + ng * ND + c0 * 16;
    float s_ap = 0.0f, s_an = 0.0f;
    #pragma unroll
    for (int k = 0; k < 16; ++k) {
        const float e  = ea[k];
        const float dp = e - ep[k] + TEPS;
        const float dn = e - en[k] + TEPS;
        s_ap += dp * dp;
        s_an += dn * dn;
    }
    #pragma unroll
    for (int m = 1; m < 16; m <<= 1) {
        s_ap += __shfl_xor(s_ap, m, 16);
        s_an += __shfl_xor(s_an, m, 16);
    }
    if (c0 == 0) {
        const float loss = fmaxf(sqrtf(s_ap) - sqrtf(s_an) + TMARGIN, 0.0f);
        lsum[a_id] = valid ? loss : 0.0f;
        lcnt[a_id] = valid ? 1.0f : 0.0f;
    }
    __syncthreads();
    if (t == 0) {
        float s = 0.0f, cc = 0.0f;
        for (int i = 0; i < 16; ++i) { s += lsum[i]; cc += lcnt[i]; }
        partial[blockIdx.x * 2]     = s;
        partial[blockIdx.x * 2 + 1] = cc;
    }
}

// ---------------- kernel 3: deterministic fixed-order final reduction ----------------
__global__ void finalize_kernel(const float* __restrict__ partial, float* __restrict__ out) {
    if (threadIdx.x == 0 && blockIdx.x == 0) {
        float s = 0.0f, c = 0.0f;
        for (int i = 0; i < 256; ++i) { s += partial[2 * i]; c += partial[2 * i + 1]; }
        out[0] = (c > 0.0f) ? s / fmaxf(c, 1.0f) : 0.0f;
    }
}

extern "C" void kernel_launch(void* const* d_in, const int* in_sizes, int n_in,
                              void* d_out, int out_size, void* d_ws, size_t ws_size,
                              hipStream_t stream) {
    const float* emb    = (const float*)d_in[0];   // [4096, 256] fp32 (normalized)
    const int*   labels = (const int*)d_in[1];     // [4096] int32
    float* ws      = (float*)d_ws;
    float* sq      = ws;                           // 4096 floats
    float* partial = ws + NB;                      // 512 floats
    float* out     = (float*)d_out;

    const size_t shmem = (size_t)(16 * DSTRIDE + 32) * sizeof(float);  // ~257 KB (<= 320 KB WGP LDS)
    hipFuncSetAttribute(reinterpret_cast<const void*>(triplet_kernel),
                        hipFuncAttributeMaxDynamicSharedMemorySize, (int)shmem);

    sq_norm_kernel<<<NB / 8, 256, 0, stream>>>(emb, sq);
    triplet_kernel<<<NB / 16, 256, shmem, stream>>>(emb, labels, sq, partial);
    finalize_kernel<<<1, 64, 0, stream>>>(partial, out);
}